// GCN_71038759076272
// MI455X (gfx1250) — compile-verified
//
#include <hip/hip_runtime.h>
#include <hip/hip_bf16.h>

// ---------------------------------------------------------------------------
// Problem constants (match the reference)
// ---------------------------------------------------------------------------
#define NU_   40000
#define NI_   60000
#define NTOT  100000
#define DF_   1024
#define DL_   128
#define DI_   64
#define NE_   1600000
#define SLOPE 0.01f
#define EPSN  1e-12f

typedef __attribute__((ext_vector_type(2))) float v2f;
typedef __attribute__((ext_vector_type(8))) float v8f;

__device__ __forceinline__ float leaky_f(float v) {
    return v > 0.0f ? v : SLOPE * v;
}

// ---------------------------------------------------------------------------
// WMMA fp32 GEMM:  C[M,N] = act( A[M,K] * op(B) + bias + addPre ) + addPost
//   op(B): TRANSB ? B is [N,K] row-major (use B[n*K+k]) : B is [K,N] (B[k*N+n])
//   LEAKY_A: apply leaky to A elements on load (fused activation of segsum)
//   LEAKY_OUT: act = leaky, else identity
//
// Block = 4 waves (128 threads). Each wave computes a 16(M) x (NT*16)(N) tile
// with NT v8f accumulators. B is staged cooperatively into LDS in 64-wide
// k-chunks, pre-swizzled so each lane reads its (b0,b1) with one ds_load_b64.
// Per k-step, all NT B operands are fetched into distinct registers first so
// the ds_loads issue back-to-back (one dscnt wait), then the NT WMMAs burst.
// A loads are software-pipelined one k-step ahead. Out-of-range M-tiles clamp
// their A row (no early return -> barriers stay uniform) and skip the store.
// Requires: N == gridDim.x * NT * 16, K % 64 == 0 (all call sites satisfy).
// ---------------------------------------------------------------------------
template <int NT, bool TRANSB, bool LEAKY_A, bool LEAKY_OUT>
__global__ void wmma_gemm_f32(const float* __restrict__ A,
                              const float* __restrict__ B,
                              const float* __restrict__ bias,     // [N] or null
                              const float* __restrict__ addPre,   // [M,N] or null
                              const float* __restrict__ addPost,  // [M,N] or null
                              float* __restrict__ C,
                              int M, int N, int K) {
    constexpr int KC    = 64;        // k-values per LDS chunk
    constexpr int STEPS = KC / 4;    // WMMA k-steps per chunk
    __shared__ v2f sB[STEPS * NT * 32];  // NT=8 -> 32 KB, NT=4 -> 16 KB

    const int tid  = threadIdx.x;
    const int lane = tid & 31;
    const int wave = tid >> 5;
    const int half = lane >> 4;      // 0: K pair {0,1}, 1: K pair {2,3}
    const int l16  = lane & 15;

    const int  mtile   = blockIdx.y * 4 + wave;
    const bool valid   = (mtile * 16) < M;
    const int  colbase = blockIdx.x * (NT * 16);

    // Clamp invalid waves onto the last tile (harmless reads, store skipped).
    const int arowIdx = valid ? (mtile * 16 + l16) : (M - 16 + l16);
    const float* __restrict__ arow = A + (size_t)arowIdx * K;

    v8f acc[NT];
#pragma unroll
    for (int t = 0; t < NT; ++t)
#pragma unroll
        for (int r = 0; r < 8; ++r) acc[t][r] = 0.0f;

    constexpr int ENT = STEPS * NT * 32;   // float2 entries per chunk

    for (int kbase = 0; kbase < K; kbase += KC) {
        // ---- cooperative B staging: sB[(kk*NT + t)*32 + ln] = B(k,col) pair
        for (int e = tid; e < ENT; e += 128) {
            const int kk  = e / (NT * 32);
            const int rem = e - kk * (NT * 32);
            const int t   = rem >> 5;
            const int ln  = rem & 31;
            const int krow = kbase + kk * 4 + ((ln >> 4) << 1);
            const int col  = colbase + t * 16 + (ln & 15);
            v2f bv;
            if (TRANSB) {
                const float* __restrict__ bc = B + (size_t)col * K + krow;
                bv[0] = bc[0]; bv[1] = bc[1];
            } else {
                bv[0] = B[(size_t)krow * N + col];
                bv[1] = B[(size_t)(krow + 1) * N + col];
            }
            sB[e] = bv;
        }
        __syncthreads();

        // ---- compute, A pipelined one k-step ahead, B batched per k-step
        v2f av;
        {
            const int ka = kbase + half * 2;
            av[0] = arow[ka]; av[1] = arow[ka + 1];
        }
#pragma unroll
        for (int kk = 0; kk < STEPS; ++kk) {
            v2f a_cur = av;
            if (kk + 1 < STEPS) {
                const int ka = kbase + (kk + 1) * 4 + half * 2;
                av[0] = arow[ka]; av[1] = arow[ka + 1];
            }
            if (LEAKY_A) { a_cur[0] = leaky_f(a_cur[0]); a_cur[1] = leaky_f(a_cur[1]); }

            // Fetch all NT B operands first (distinct regs -> one dscnt wait).
            v2f bvs[NT];
#pragma unroll
            for (int t = 0; t < NT; ++t)
                bvs[t] = sB[(kk * NT + t) * 32 + lane];

#pragma unroll
            for (int t = 0; t < NT; ++t)
                acc[t] = __builtin_amdgcn_wmma_f32_16x16x4_f32(
                    false, a_cur, false, bvs[t], (short)0, acc[t], false, false);
        }
        __syncthreads();
    }

    if (!valid) return;

    // epilogue: VGPR r -> row (tile + half*8 + r), lane l16 -> col
    const int rowbase = mtile * 16 + half * 8;
#pragma unroll
    for (int t = 0; t < NT; ++t) {
        const int col = colbase + t * 16 + l16;
        const float bs = bias ? bias[col] : 0.0f;
#pragma unroll
        for (int r = 0; r < 8; ++r) {
            const size_t idx = (size_t)(rowbase + r) * N + col;
            float v = acc[t][r] + bs;
            if (addPre)  v += addPre[idx];
            if (LEAKY_OUT) v = leaky_f(v);
            if (addPost) v += addPost[idx];
            C[idx] = v;
        }
    }
}

// ---------------------------------------------------------------------------
// concat(preference, temp) + row-wise L2 normalize over DL=128 columns.
// One wave per row; each lane owns a float4.
// ---------------------------------------------------------------------------
__global__ void concat_normalize(const float* __restrict__ pref,
                                 float* __restrict__ xbuf) {
    const int wavesPerBlock = blockDim.x >> 5;
    const int row  = blockIdx.x * wavesPerBlock + (threadIdx.x >> 5);
    const int lane = threadIdx.x & 31;
    if (row >= NTOT) return;

    const float* srcp = (row < NU_) ? (pref + (size_t)row * DL_)
                                    : (xbuf + (size_t)row * DL_);
    const float4 v = *reinterpret_cast<const float4*>(srcp + lane * 4);
    float ss = v.x * v.x + v.y * v.y + v.z * v.z + v.w * v.w;
#pragma unroll
    for (int off = 16; off > 0; off >>= 1) ss += __shfl_xor(ss, off, 32);
    const float inv = 1.0f / fmaxf(sqrtf(ss), EPSN);
    float4 o; o.x = v.x * inv; o.y = v.y * inv; o.z = v.z * inv; o.w = v.w * inv;
    *reinterpret_cast<float4*>(xbuf + (size_t)row * DL_ + lane * 4) = o;
}

// ---------------------------------------------------------------------------
// Zero fill (graph-replay safe re-init of accumulation buffers)
// ---------------------------------------------------------------------------
__global__ void zero_f32(float* __restrict__ p, size_t n) {
    size_t i = (size_t)blockIdx.x * blockDim.x + threadIdx.x;
    const size_t stride = (size_t)gridDim.x * blockDim.x;
    for (; i < n; i += stride) p[i] = 0.0f;
}

// ---------------------------------------------------------------------------
// Edge scatter-add: h[dst] += xw[src].  One wave per edge; each lane owns
// D/32 consecutive floats (float4 for D=128, float2 for D=64).
// ---------------------------------------------------------------------------
template <int D>
__global__ void scatter_add(const float* __restrict__ xw,
                            const int* __restrict__ src,
                            const int* __restrict__ dst,
                            float* __restrict__ h, int ne) {
    constexpr int VEC = D / 32;
    const int wavesPerBlock = blockDim.x >> 5;
    const int lane = threadIdx.x & 31;
    int wid = blockIdx.x * wavesPerBlock + (threadIdx.x >> 5);
    const int totalWaves = gridDim.x * wavesPerBlock;

    for (int e = wid; e < ne; e += totalWaves) {
        const int s = src[e];
        const int d = dst[e];
        const float* __restrict__ xr = xw + (size_t)s * D + lane * VEC;
        float* __restrict__ hr = h + (size_t)d * D + lane * VEC;
        if constexpr (VEC == 4) {
            const float4 v = *reinterpret_cast<const float4*>(xr);
            atomicAdd(hr + 0, v.x);
            atomicAdd(hr + 1, v.y);
            atomicAdd(hr + 2, v.z);
            atomicAdd(hr + 3, v.w);
        } else {
            const float2 v = *reinterpret_cast<const float2*>(xr);
            atomicAdd(hr + 0, v.x);
            atomicAdd(hr + 1, v.y);
        }
    }
}

// ---------------------------------------------------------------------------
// Orchestration
// ---------------------------------------------------------------------------
extern "C" void kernel_launch(void* const* d_in, const int* in_sizes, int n_in,
                              void* d_out, int out_size, void* d_ws, size_t ws_size,
                              hipStream_t stream) {
    const float* features = (const float*)d_in[0];   // [NI, DF]
    const float* id_emb   = (const float*)d_in[1];   // [N, DI]
    const float* pref     = (const float*)d_in[2];   // [NU, DL]
    const float* mlp_w    = (const float*)d_in[3];   // [DL, DF]
    const float* mlp_b    = (const float*)d_in[4];   // [DL]
    const float* conv1_w  = (const float*)d_in[5];   // [DL, DL]
    const float* lin1_w   = (const float*)d_in[6];   // [DI, DL]
    const float* lin1_b   = (const float*)d_in[7];   // [DI]
    const float* g1_w     = (const float*)d_in[8];   // [DI, DL]
    const float* g1_b     = (const float*)d_in[9];   // [DI]
    const float* conv2_w  = (const float*)d_in[10];  // [DI, DI]
    const float* lin2_w   = (const float*)d_in[11];  // [DI, DI]
    const float* lin2_b   = (const float*)d_in[12];  // [DI]
    const float* g2_w     = (const float*)d_in[13];  // [DI, DI]
    const float* g2_b     = (const float*)d_in[14];  // [DI]
    const int*   eidx     = (const int*)d_in[15];    // [2, E]
    const int*   esrc     = eidx;
    const int*   edst     = eidx + NE_;
    float*       out      = (float*)d_out;           // [N, DI]

    // Workspace layout (floats). x1 overwrites xbuf; layer 2 reuses xw/hraw/xhat.
    float* xbuf = (float*)d_ws;                       // [N,128] -> later x1 [N,64]
    float* xw   = xbuf + (size_t)NTOT * DL_;          // [N,128] / [N,64]
    float* hraw = xw   + (size_t)NTOT * DL_;          // [N,128] / [N,64]
    float* xhat = hraw + (size_t)NTOT * DL_;          // [N,64]
    (void)ws_size; (void)in_sizes; (void)n_in; (void)out_size;

    const dim3 gemmBlk(128);  // 4 waves, 4 M-tiles per block
    auto mblocks = [](int M) { return (M / 16 + 3) / 4; };

    // ---- MLP: temp = features @ mlp_w^T + mlp_b  -> xbuf rows NU..N-1
    // NT=8: full 128-wide output per wave, A read exactly once.
    wmma_gemm_f32<8, true, false, false><<<dim3(1, mblocks(NI_)), gemmBlk, 0, stream>>>(
        features, mlp_w, mlp_b, nullptr, nullptr,
        xbuf + (size_t)NU_ * DL_, NI_, DL_, DF_);

    // ---- concat + L2 normalize -> xbuf [N,128]
    concat_normalize<<<dim3((NTOT + 7) / 8), dim3(256), 0, stream>>>(pref, xbuf);

    // =========================== layer 1 ===================================
    // xw = x @ conv1_w   [N,128]
    wmma_gemm_f32<8, false, false, false><<<dim3(1, mblocks(NTOT)), gemmBlk, 0, stream>>>(
        xbuf, conv1_w, nullptr, nullptr, nullptr, xw, NTOT, DL_, DL_);

    // hraw = segment_sum(xw[src] -> dst)
    zero_f32<<<dim3(2048), dim3(256), 0, stream>>>(hraw, (size_t)NTOT * DL_);
    scatter_add<DL_><<<dim3(4096), dim3(256), 0, stream>>>(xw, esrc, edst, hraw, NE_);

    // xhat = leaky(x @ lin1_w^T + lin1_b) + id_emb   [N,64]
    wmma_gemm_f32<4, true, false, true><<<dim3(1, mblocks(NTOT)), gemmBlk, 0, stream>>>(
        xbuf, lin1_w, lin1_b, nullptr, id_emb, xhat, NTOT, DI_, DL_);

    // x1 = leaky( leaky(hraw) @ g1_w^T + g1_b + xhat )  -> xbuf [N,64]
    wmma_gemm_f32<4, true, true, true><<<dim3(1, mblocks(NTOT)), gemmBlk, 0, stream>>>(
        hraw, g1_w, g1_b, xhat, nullptr, xbuf, NTOT, DI_, DL_);

    // =========================== layer 2 ===================================
    // xw = x1 @ conv2_w  [N,64]
    wmma_gemm_f32<4, false, false, false><<<dim3(1, mblocks(NTOT)), gemmBlk, 0, stream>>>(
        xbuf, conv2_w, nullptr, nullptr, nullptr, xw, NTOT, DI_, DI_);

    zero_f32<<<dim3(2048), dim3(256), 0, stream>>>(hraw, (size_t)NTOT * DI_);
    scatter_add<DI_><<<dim3(4096), dim3(256), 0, stream>>>(xw, esrc, edst, hraw, NE_);

    // xhat = leaky(x1 @ lin2_w^T + lin2_b) + id_emb
    wmma_gemm_f32<4, true, false, true><<<dim3(1, mblocks(NTOT)), gemmBlk, 0, stream>>>(
        xbuf, lin2_w, lin2_b, nullptr, id_emb, xhat, NTOT, DI_, DI_);

    // out = leaky( leaky(hraw) @ g2_w^T + g2_b + xhat )
    wmma_gemm_f32<4, true, true, true><<<dim3(1, mblocks(NTOT)), gemmBlk, 0, stream>>>(
        hraw, g2_w, g2_b, xhat, nullptr, out, NTOT, DI_, DI_);
}